// masked_softmax_random_6674379178453
// MI455X (gfx1250) — compile-verified
//
#include <hip/hip_runtime.h>
#include <stdint.h>

// Problem constants (from the reference): X[4, 5843, 5843] fp32, idx[5843*100] i32.
#define QDIM   5843
#define KDIM   5843
#define TOPK   100
#define BATCH  4
#define KPAD   5888          // KDIM rounded up to a multiple of 256 (and of 4)
#define BLOCK  256           // 8 wave32s
#define NWAVES (BLOCK / 32)
#define ITER   ((KDIM + BLOCK - 1) / BLOCK)   // 23 columns per thread
#define NEG_VALUE -10000000.0f

__device__ __forceinline__ float block_reduce_max(float v, float* red) {
#pragma unroll
  for (int off = 16; off > 0; off >>= 1)
    v = fmaxf(v, __shfl_xor(v, off, 32));
  const int wave = threadIdx.x >> 5, lane = threadIdx.x & 31;
  if (lane == 0) red[wave] = v;
  __syncthreads();
  if (wave == 0) {
    float w = (lane < NWAVES) ? red[lane] : NEG_VALUE;
#pragma unroll
    for (int off = NWAVES / 2; off > 0; off >>= 1)
      w = fmaxf(w, __shfl_xor(w, off, 32));
    if (lane == 0) red[0] = w;
  }
  __syncthreads();
  float r = red[0];
  __syncthreads();              // red[] reusable afterwards
  return r;
}

__device__ __forceinline__ float block_reduce_sum(float v, float* red) {
#pragma unroll
  for (int off = 16; off > 0; off >>= 1)
    v += __shfl_xor(v, off, 32);
  const int wave = threadIdx.x >> 5, lane = threadIdx.x & 31;
  if (lane == 0) red[wave] = v;
  __syncthreads();
  if (wave == 0) {
    float w = (lane < NWAVES) ? red[lane] : 0.0f;
#pragma unroll
    for (int off = NWAVES / 2; off > 0; off >>= 1)
      w += __shfl_xor(w, off, 32);
    if (lane == 0) red[0] = w;
  }
  __syncthreads();
  float r = red[0];
  __syncthreads();
  return r;
}

// One workgroup per query row q; loops over the 4 batches. The selected-index
// set depends only on q, so the NEG_VALUE background in LDS is initialized
// once and each batch's async gather overwrites exactly the same slots.
__global__ __launch_bounds__(BLOCK)
void sparse_masked_softmax(const float* __restrict__ X,
                           const int*   __restrict__ rnd,
                           float*       __restrict__ out) {
  __shared__ float vals[KPAD];     // the materialized X_random row
  __shared__ float red[NWAVES];    // reduction scratch

  const int q = blockIdx.x;
  const int t = threadIdx.x;

  // ---- background init: vals[k] = -1e7, 128-bit LDS stores ----
  {
    float4* v4 = (float4*)vals;
    const float4 negv = make_float4(NEG_VALUE, NEG_VALUE, NEG_VALUE, NEG_VALUE);
#pragma unroll
    for (int i = t; i < KPAD / 4; i += BLOCK) v4[i] = negv;
  }

  const bool active = (t < TOPK);
  int my_idx = 0;
  if (active) my_idx = rnd[q * TOPK + t];   // one index per gather lane

  for (int b = 0; b < BATCH; ++b) {
    __syncthreads();   // background / previous batch fully consumed

    // ---- async gather: X[b,q,idx] -> LDS vals[idx]; per-lane LDS scatter,
    //      no VGPR round-trip, tracked by ASYNCcnt (CDNA5 TDM-class path) ----
    if (active) {
      const float* srcp = X + ((size_t)b * QDIM + q) * (size_t)KDIM + my_idx;
      unsigned long long ga = (unsigned long long)(uintptr_t)srcp;
      // low 32 bits of a generic LDS pointer == LDS byte offset (aperture map)
      unsigned lo = (unsigned)(uintptr_t)&vals[my_idx];
      asm volatile("global_load_async_to_lds_b32 %0, %1, off"
                   :: "v"(lo), "v"(ga) : "memory");
    }
    asm volatile("s_wait_asynccnt 0x0" ::: "memory");
    __syncthreads();   // gathered values visible to all waves

    // ---- single LDS read per column into registers; row max ----
    float x[ITER];
    float m = NEG_VALUE;
#pragma unroll
    for (int i = 0; i < ITER; ++i) {
      const int k = t + i * BLOCK;
      if (k < KDIM) { x[i] = vals[k]; m = fmaxf(m, x[i]); }
    }
    m = block_reduce_max(m, red);

    // ---- exp + sum in registers; unselected -> exp(-1e7 - m) == 0.0f,
    //      identical to the reference's underflowed denominator terms ----
    float s = 0.0f;
#pragma unroll
    for (int i = 0; i < ITER; ++i) {
      const int k = t + i * BLOCK;
      if (k < KDIM) { x[i] = __expf(x[i] - m); s += x[i]; }
    }
    s = block_reduce_sum(s, red);
    const float inv = 1.0f / s;

    // ---- dense coalesced non-temporal write (546 MB write-once) ----
    float* orow = out + ((size_t)b * QDIM + q) * (size_t)KDIM;
#pragma unroll
    for (int i = 0; i < ITER; ++i) {
      const int k = t + i * BLOCK;
      if (k < KDIM) __builtin_nontemporal_store(x[i] * inv, &orow[k]);
    }
  }
}

extern "C" void kernel_launch(void* const* d_in, const int* in_sizes, int n_in,
                              void* d_out, int out_size, void* d_ws, size_t ws_size,
                              hipStream_t stream) {
  const float* X   = (const float*)d_in[0];   // [4, 5843, 5843] fp32
  const int*   rnd = (const int*)d_in[1];     // [5843*100] i32
  float*       out = (float*)d_out;           // [4, 5843, 5843] fp32
  (void)in_sizes; (void)n_in; (void)out_size; (void)d_ws; (void)ws_size;
  hipLaunchKernelGGL(sparse_masked_softmax, dim3(QDIM), dim3(BLOCK), 0, stream,
                     X, rnd, out);
}